// Hippocampus_45724221833469
// MI455X (gfx1250) — compile-verified
//
#include <hip/hip_runtime.h>

typedef __attribute__((ext_vector_type(2))) float v2f;
typedef __attribute__((ext_vector_type(4))) float v4f;
typedef __attribute__((ext_vector_type(8))) float v8f;

#define WMMA_F32(a, b, c) \
    __builtin_amdgcn_wmma_f32_16x16x4_f32(false, (a), false, (b), (short)0, (c), false, false)

// ---------------------------------------------------------------------------
// Generic row-major f32 GEMM with bias: C[M,N] = A[M,K] @ B[K,N] + bias[N]
// One wave computes a 16x64 tile (A frag reused across 4 B frags).
// Requires M%16==0, N%64==0, K%4==0.
// ---------------------------------------------------------------------------
__global__ __launch_bounds__(256) void gemm_bias_f32(
    const float* __restrict__ A, const float* __restrict__ B,
    const float* __restrict__ bias, float* __restrict__ C,
    int M, int N, int K)
{
    const int lane = threadIdx.x & 31;
    const int wave = (int)((blockIdx.x * blockDim.x + threadIdx.x) >> 5);
    const int colQuads = N >> 6;               // number of 64-wide column tiles
    const int rt = wave / colQuads;            // 16-row tile index
    const int cq = wave % colQuads;            // 64-col tile index
    if (rt * 16 >= M) return;

    const int half = lane >> 4;                // 0 or 1 (K sub-lane group)
    const int lm   = lane & 15;
    const int row  = rt * 16 + lm;             // A-frag row (M = lane%16)
    const int n0   = cq * 64 + lm;             // B-frag col (N = lane%16)

    v8f acc0 = {}, acc1 = {}, acc2 = {}, acc3 = {};

    for (int kk = 0; kk < K; kk += 4) {
        const int k = kk + 2 * half;
        // A frag: lane holds A[row][k], A[row][k+1]
        v2f a = *(const v2f*)(A + (size_t)row * K + k);
        const float* Bk = B + (size_t)k * N;
        v2f b0 = { Bk[n0 +  0], Bk[(size_t)N + n0 +  0] };
        v2f b1 = { Bk[n0 + 16], Bk[(size_t)N + n0 + 16] };
        v2f b2 = { Bk[n0 + 32], Bk[(size_t)N + n0 + 32] };
        v2f b3 = { Bk[n0 + 48], Bk[(size_t)N + n0 + 48] };
        acc0 = WMMA_F32(a, b0, acc0);
        acc1 = WMMA_F32(a, b1, acc1);
        acc2 = WMMA_F32(a, b2, acc2);
        acc3 = WMMA_F32(a, b3, acc3);
    }

    const float bb0 = bias[n0 +  0];
    const float bb1 = bias[n0 + 16];
    const float bb2 = bias[n0 + 32];
    const float bb3 = bias[n0 + 48];

#pragma unroll
    for (int e = 0; e < 8; ++e) {
        const int r = rt * 16 + e + 8 * half;  // C/D layout: VGPR e -> rows e, e+8
        float* Crow = C + (size_t)r * N;
        Crow[n0 +  0] = acc0[e] + bb0;
        Crow[n0 + 16] = acc1[e] + bb1;
        Crow[n0 + 32] = acc2[e] + bb2;
        Crow[n0 + 48] = acc3[e] + bb3;
    }
}

// ---------------------------------------------------------------------------
// Scores + grouped softmax, 4 consecutive units per workgroup.
// grid = (128 u-quads, 16 b-tiles), block = 256 (8 waves), 64 KB LDS.
// S[uu][bl][m] = sum_d attn[b,u,d]*memA[m,u,d] * temp[u]; diag -> -inf;
// grouped softmax over m1 (m = m1*8+g, m1 in [0,32)); /8;
// then coalesced b128 writes: weights[b][m][u0..u0+3].
// ---------------------------------------------------------------------------
__global__ __launch_bounds__(256) void scores_softmax(
    const float* __restrict__ attn,   // [256,512,16]
    const float* __restrict__ memA,   // [256,512,16]
    const float* __restrict__ temp,   // [512]
    float* __restrict__ weights)      // [256,256,512]
{
    const int u0   = blockIdx.x * 4;  // first of 4 consecutive units
    const int bt   = blockIdx.y;      // 0..15
    const int lane = threadIdx.x & 31;
    const int wv   = threadIdx.x >> 5;
    const int half = lane >> 4;
    const int lm   = lane & 15;

    __shared__ float S[4][16][256];   // 64 KB

    for (int uu = 0; uu < 4; ++uu) {
        const int u = u0 + uu;
        const float tscale = temp[u];

        // A fragments (b rows, K = d): attn[(b*512+u)*16 + d]
        const float* Arow = attn + ((size_t)(bt * 16 + lm) * 512 + u) * 16;
        v2f a0 = *(const v2f*)(Arow +  0 + 2 * half);
        v2f a1 = *(const v2f*)(Arow +  4 + 2 * half);
        v2f a2 = *(const v2f*)(Arow +  8 + 2 * half);
        v2f a3 = *(const v2f*)(Arow + 12 + 2 * half);

#pragma unroll
        for (int t = 0; t < 2; ++t) {
            const int mt = wv * 2 + t;        // m-tile 0..15
            const int m  = mt * 16 + lm;      // B-frag col (N = m)
            // B fragment: B[k=d][n=m] = memA[m][u][d]
            const float* Brow = memA + ((size_t)m * 512 + u) * 16;
            v2f b0 = *(const v2f*)(Brow +  0 + 2 * half);
            v2f b1 = *(const v2f*)(Brow +  4 + 2 * half);
            v2f b2 = *(const v2f*)(Brow +  8 + 2 * half);
            v2f b3 = *(const v2f*)(Brow + 12 + 2 * half);

            v8f acc = {};
            acc = WMMA_F32(a0, b0, acc);
            acc = WMMA_F32(a1, b1, acc);
            acc = WMMA_F32(a2, b2, acc);
            acc = WMMA_F32(a3, b3, acc);

#pragma unroll
            for (int e = 0; e < 8; ++e) {
                const int bl = e + 8 * half;                   // local b row
                float v = acc[e] * tscale;
                if (bt * 16 + bl == m) v = -__builtin_inff();  // eye mask
                S[uu][bl][m] = v;
            }
        }
    }
    __syncthreads();

    // Grouped softmax: 512 (uu, bl, g) tasks over 256 threads.
    for (int task = threadIdx.x; task < 512; task += 256) {
        const int uu = task >> 7;
        const int bl = (task >> 3) & 15;
        const int g  = task & 7;
        float mx = -__builtin_inff();
#pragma unroll
        for (int m1 = 0; m1 < 32; ++m1)
            mx = fmaxf(mx, S[uu][bl][m1 * 8 + g]);
        float sum = 0.f;
#pragma unroll
        for (int m1 = 0; m1 < 32; ++m1) {
            float e = __expf(S[uu][bl][m1 * 8 + g] - mx);  // exp(-inf)=0 handles mask
            S[uu][bl][m1 * 8 + g] = e;
            sum += e;
        }
        const float inv = 1.0f / (8.0f * sum);              // softmax + /8
#pragma unroll
        for (int m1 = 0; m1 < 32; ++m1)
            S[uu][bl][m1 * 8 + g] *= inv;
    }
    __syncthreads();

    // Write weights[b][m][u0..u0+3] as 16-byte stores (u fastest axis).
    for (int i = threadIdx.x; i < 16 * 256; i += 256) {
        const int bl = i >> 8;
        const int m  = i & 255;
        v4f w = { S[0][bl][m], S[1][bl][m], S[2][bl][m], S[3][bl][m] };
        *(v4f*)(weights + ((size_t)(bt * 16 + bl) * 256 + m) * 512 + u0) = w;
    }
}

// ---------------------------------------------------------------------------
// outputs[b,u,d] = sum_m weights[b,m,u] * memV[m,u,d]
// One wave computes 16x16 tiles for 4 consecutive units (shared b128 A loads).
// 2048 waves = 16 b-tiles * 128 u-quads -> 256 blocks of 8 waves.
// ---------------------------------------------------------------------------
__global__ __launch_bounds__(256) void out_gemm(
    const float* __restrict__ weights, // [256,256,512]
    const float* __restrict__ memV,    // [256,512,16]
    float* __restrict__ outp)          // [256,512,16]
{
    const int gwave = (int)((blockIdx.x * blockDim.x + threadIdx.x) >> 5);
    const int u0 = (gwave & 127) * 4;  // 4 consecutive units
    const int bt = gwave >> 7;         // 0..15
    const int lane = threadIdx.x & 31;
    const int half = lane >> 4;
    const int lm   = lane & 15;
    const int b    = bt * 16 + lm;     // A-frag row

    v8f acc0 = {}, acc1 = {}, acc2 = {}, acc3 = {};

    for (int kk = 0; kk < 256; kk += 4) {
        const int m = kk + 2 * half;
        // A: weights[b][m][u0..u0+3] and weights[b][m+1][u0..u0+3] (b128 each)
        const float* Ap = weights + ((size_t)b * 256 + m) * 512 + u0;
        v4f p0 = *(const v4f*)(Ap);
        v4f p1 = *(const v4f*)(Ap + 512);
        v2f a0 = { p0.x, p1.x };
        v2f a1 = { p0.y, p1.y };
        v2f a2 = { p0.z, p1.z };
        v2f a3 = { p0.w, p1.w };
        // B: memV[m][u][d], d = lane%16; u offsets are 16-float immediates
        const float* Bp = memV + ((size_t)m * 512 + u0) * 16 + lm;
        v2f bf0 = { Bp[ 0], Bp[512 * 16 +  0] };
        v2f bf1 = { Bp[16], Bp[512 * 16 + 16] };
        v2f bf2 = { Bp[32], Bp[512 * 16 + 32] };
        v2f bf3 = { Bp[48], Bp[512 * 16 + 48] };
        acc0 = WMMA_F32(a0, bf0, acc0);
        acc1 = WMMA_F32(a1, bf1, acc1);
        acc2 = WMMA_F32(a2, bf2, acc2);
        acc3 = WMMA_F32(a3, bf3, acc3);
    }

#pragma unroll
    for (int e = 0; e < 8; ++e) {
        const int bb = bt * 16 + e + 8 * half;
        float* orow = outp + ((size_t)bb * 512 + u0) * 16 + lm;
        orow[ 0] = acc0[e];
        orow[16] = acc1[e];
        orow[32] = acc2[e];
        orow[48] = acc3[e];
    }
}

// ---------------------------------------------------------------------------
// Launch
// ---------------------------------------------------------------------------
extern "C" void kernel_launch(void* const* d_in, const int* in_sizes, int n_in,
                              void* d_out, int out_size, void* d_ws, size_t ws_size,
                              hipStream_t stream) {
    const float* x    = (const float*)d_in[0];   // [256,1024]
    const float* W1   = (const float*)d_in[1];   // [1024,1024]
    const float* b1   = (const float*)d_in[2];   // [1024]
    const float* W2   = (const float*)d_in[3];   // [1024,8192]
    const float* b2   = (const float*)d_in[4];   // [8192]
    const float* temp = (const float*)d_in[5];   // [512]
    const float* memA = (const float*)d_in[6];   // [256,512,16]
    const float* memV = (const float*)d_in[7];   // [256,512,16]

    float* out       = (float*)d_out;
    float* attention = out;                       // [256,512,16]  = 2,097,152
    float* weights   = out + 2097152;             // [256,256,512] = 33,554,432
    float* outputs   = out + 2097152 + 33554432;  // [256,512,16]

    float* h = (float*)d_ws;                      // [256,1024] scratch (1 MB)

    // GEMM1: h = x @ W1 + b1   (16x64 tiles: 16*16 = 256 waves -> 32 blocks)
    gemm_bias_f32<<<32, 256, 0, stream>>>(x, W1, b1, h, 256, 1024, 1024);

    // GEMM2: attention = h @ W2 + b2   (16*128 = 2048 waves -> 256 blocks)
    gemm_bias_f32<<<256, 256, 0, stream>>>(h, W2, b2, attention, 256, 8192, 1024);

    // Scores + grouped softmax -> weights (4 units per workgroup)
    scores_softmax<<<dim3(128, 16), 256, 0, stream>>>(attention, memA, temp, weights);

    // outputs = weights @ memV   (2048 waves -> 256 blocks)
    out_gemm<<<256, 256, 0, stream>>>(weights, memV, outputs);
}